// ProposalProcess_52329881534482
// MI455X (gfx1250) — compile-verified
//
#include <hip/hip_runtime.h>
#include <hip/hip_bf16.h>
#include <stdint.h>

#define B_IMG 64
#define Q_NUM 900
#define C_NUM 91
#define QC_NUM (Q_NUM * C_NUM)
#define NMS_PRE_N 1000
#define KPAD 1024
#define NUM_SELECT_N 100
#define MIN_SCORE_F 0.05f
#define NMS_THR_F 0.7f

typedef unsigned int uint32x4 __attribute__((ext_vector_type(4)));
typedef int int32x4 __attribute__((ext_vector_type(4)));
typedef int int32x8 __attribute__((ext_vector_type(8)));

// CDNA5 Tensor Data Mover (device pass only).
#if defined(__has_builtin)
#if __has_builtin(__builtin_amdgcn_tensor_load_to_lds) && __has_builtin(__builtin_amdgcn_s_wait_tensorcnt)
#define USE_TDM 1
#endif
#if __has_builtin(__builtin_amdgcn_global_load_async_to_lds_b32) && __has_builtin(__builtin_amdgcn_s_wait_asynccnt)
#define USE_ASYNC_LDS 1
#endif
#endif
#ifndef USE_TDM
#define USE_TDM 0
#endif
#ifndef USE_ASYNC_LDS
#define USE_ASYNC_LDS 0
#endif

#define GAS __attribute__((address_space(1)))
#define LAS __attribute__((address_space(3)))

// ---- order-preserving key helpers -----------------------------------------
// dkey: ascending dkey order == descending float order (exact, invertible)
__device__ __forceinline__ unsigned int f32_to_dkey(float f) {
  unsigned int u = __float_as_uint(f);
  unsigned int k = u ^ ((unsigned int)((int)u >> 31) | 0x80000000u);
  return ~k;
}
__device__ __forceinline__ float dkey_to_f32(unsigned int dkey) {
  unsigned int k = ~dkey;
  unsigned int u = (k & 0x80000000u) ? (k ^ 0x80000000u) : ~k;
  return __uint_as_float(u);
}
__device__ __forceinline__ unsigned int score_dkey(float logit) {
  float sc = 1.0f / (1.0f + expf(-logit));
  float s = (sc > MIN_SCORE_F) ? sc : -1.0f;
  return f32_to_dkey(s);
}

// ---- 1024-element LDS bitonic sort (ascending), any blockDim ---------------
__device__ __forceinline__ void bitonic_sort_1024(unsigned long long* keys) {
  for (unsigned int k = 2; k <= KPAD; k <<= 1) {
    for (unsigned int j = k >> 1; j > 0; j >>= 1) {
      for (unsigned int i = threadIdx.x; i < KPAD; i += blockDim.x) {
        unsigned int ixj = i ^ j;
        if (ixj > i) {
          unsigned long long a = keys[i];
          unsigned long long c = keys[ixj];
          bool up = ((i & k) == 0);
          if ((a > c) == up) { keys[i] = c; keys[ixj] = a; }
        }
      }
      __syncthreads();
    }
  }
}

// ===========================================================================
// Kernel 0: grid-wide key precompute (sigmoid + mask -> monotone u32 key)
// ===========================================================================
__global__ __launch_bounds__(256)
void score_key_kernel(const float* __restrict__ logits,
                      unsigned int* __restrict__ keys, int n) {
  int stride = (int)(gridDim.x * blockDim.x);
  for (int i = (int)(blockIdx.x * blockDim.x + threadIdx.x); i < n; i += stride) {
    __builtin_prefetch(logits + i + 2 * stride, 0, 0);
    keys[i] = score_dkey(logits[i]);
  }
}

// ===========================================================================
// Kernel 1: per-image exact top-1000 (8-pass MSB radix-select over unique
// 64-bit (dkey<<32|idx) keys, then LDS bitonic) + box scaling (SoA output)
// ===========================================================================
template <bool HAVE_KEYS>
__global__ __launch_bounds__(256)
void topk_select_kernel(const float* __restrict__ logits,
                        const unsigned int* __restrict__ pre_keys,
                        const float* __restrict__ boxes_in,
                        const int* __restrict__ tsizes,
                        float* __restrict__ ws_tops,
                        int* __restrict__ ws_label,
                        float* __restrict__ ws_boxes) {
  const int b = blockIdx.x;
  const float* lg = logits + (size_t)b * QC_NUM;
  const unsigned int* pk = pre_keys + (size_t)b * QC_NUM;

  __shared__ unsigned int hist[256];
  __shared__ unsigned long long sh_prefix;
  __shared__ unsigned int sh_rem;
  __shared__ unsigned int sh_cnt;
  __shared__ unsigned long long keys[KPAD];

  if (threadIdx.x == 0) { sh_prefix = 0ull; sh_rem = NMS_PRE_N; sh_cnt = 0u; }
  __syncthreads();

  unsigned long long pmask = 0ull;
  for (int shift = 56; shift >= 0; shift -= 8) {
    for (unsigned int i = threadIdx.x; i < 256u; i += blockDim.x) hist[i] = 0u;
    __syncthreads();
    const unsigned long long pref = sh_prefix;
    for (unsigned int e = threadIdx.x; e < QC_NUM; e += blockDim.x) {
      unsigned long long k;
      if (HAVE_KEYS) {
        __builtin_prefetch(pk + e + 4u * blockDim.x, 0, 0);
        k = ((unsigned long long)pk[e] << 32) | (unsigned long long)e;
      } else {
        __builtin_prefetch(lg + e + 4u * blockDim.x, 0, 0);
        k = ((unsigned long long)score_dkey(lg[e]) << 32) | (unsigned long long)e;
      }
      if ((k & pmask) == pref)
        atomicAdd(&hist[(unsigned int)(k >> shift) & 0xFFu], 1u);
    }
    __syncthreads();
    if (threadIdx.x == 0) {
      unsigned int rem = sh_rem, cum = 0u; int d = 255;
      for (int i = 0; i < 256; ++i) {
        unsigned int c = hist[i];
        if (cum + c >= rem) { d = i; break; }
        cum += c;
      }
      sh_rem = rem - cum;
      sh_prefix = pref | ((unsigned long long)d << shift);
    }
    __syncthreads();
    pmask |= (0xFFull << shift);
  }

  const unsigned long long T = sh_prefix;  // exactly 1000 keys are <= T
  for (unsigned int e = threadIdx.x; e < QC_NUM; e += blockDim.x) {
    unsigned long long k;
    if (HAVE_KEYS) k = ((unsigned long long)pk[e] << 32) | (unsigned long long)e;
    else           k = ((unsigned long long)score_dkey(lg[e]) << 32) | (unsigned long long)e;
    if (k <= T) {
      unsigned int p = atomicAdd(&sh_cnt, 1u);
      if (p < KPAD) keys[p] = k;
    }
  }
  __syncthreads();
  unsigned int cnt = sh_cnt; if (cnt > KPAD) cnt = KPAD;
  for (unsigned int i = threadIdx.x; i < KPAD; i += blockDim.x)
    if (i >= cnt) keys[i] = ~0ull;
  __syncthreads();

  bitonic_sort_1024(keys);

  const float W = (float)tsizes[b * 2 + 1];
  const float H = (float)tsizes[b * 2 + 0];
  float* wsb = ws_boxes + (size_t)b * 4 * NMS_PRE_N;  // SoA: [4][1000] per image
  for (unsigned int i = threadIdx.x; i < NMS_PRE_N; i += blockDim.x) {
    unsigned long long k = keys[i];
    size_t o = (size_t)b * NMS_PRE_N + i;
    if (k == ~0ull) {  // defensive (cannot occur: 81900 > 1000)
      ws_tops[o] = -1.0f; ws_label[o] = 0;
      wsb[0 * NMS_PRE_N + i] = 0.f; wsb[1 * NMS_PRE_N + i] = 0.f;
      wsb[2 * NMS_PRE_N + i] = 0.f; wsb[3 * NMS_PRE_N + i] = 0.f;
      continue;
    }
    float val = dkey_to_f32((unsigned int)(k >> 32));
    unsigned int flat = (unsigned int)(k & 0xFFFFFFFFull);
    int lbl = (int)(flat % C_NUM);
    int q   = (int)(flat / C_NUM);
    const float* bb = boxes_in + ((size_t)b * Q_NUM + (size_t)q) * 4;
    float cx = bb[0], cy = bb[1], w = bb[2], h = bb[3];
    ws_tops[o] = val;
    ws_label[o] = lbl;
    wsb[0 * NMS_PRE_N + i] = (cx - 0.5f * w) * W;
    wsb[1 * NMS_PRE_N + i] = (cy - 0.5f * h) * H;
    wsb[2 * NMS_PRE_N + i] = (cx + 0.5f * w) * W;
    wsb[3 * NMS_PRE_N + i] = (cy + 0.5f * h) * H;
  }
}

// ===========================================================================
// Kernel 2: TDM boxes(SoA) -> LDS, async scores -> LDS, greedy NMS scan,
// final top-100 bitonic, write (boxes, scores, labels)
// ===========================================================================
__global__ __launch_bounds__(1024)
void nms_finalize_kernel(const float* __restrict__ ws_tops,
                         const int* __restrict__ ws_label,
                         const float* __restrict__ ws_boxes,
                         float* __restrict__ out) {
  const int b = blockIdx.x;
  __shared__ __align__(16) float bxs[4][NMS_PRE_N];  // SoA 16000 B tile
  __shared__ __align__(16) float tops[NMS_PRE_N];
  __shared__ float area[NMS_PRE_N];
  __shared__ unsigned char supp[NMS_PRE_N];
  __shared__ unsigned char keep[NMS_PRE_N];
  __shared__ int sh_keep;
  __shared__ unsigned long long keys[KPAD];

  const float* gbx = ws_boxes + (size_t)b * 4 * NMS_PRE_N;

#if USE_TDM
  if (threadIdx.x < 32) {  // wave 0 issues one TDM DMA of the 16KB SoA tile
    unsigned long long ga = (unsigned long long)(uintptr_t)gbx;
    unsigned int lds = (unsigned int)(uintptr_t)(&bxs[0][0]);  // addr[31:0] == LDS offset
    uint32x4 g0;
    g0[0] = 1u;                                             // count=1, user descriptor
    g0[1] = lds;                                            // lds_addr [63:32]
    g0[2] = (unsigned int)(ga & 0xFFFFFFFFull);             // global_addr[31:0]
    g0[3] = (unsigned int)((ga >> 32) & 0x01FFFFFFull) | (2u << 30);  // addr[56:32] | type=2
    const unsigned long long n8 = (unsigned long long)(4 * NMS_PRE_N * 4) / 8ull;  // 2000 x 8B
    unsigned long long qw0 = (3ull << 16) | ((n8 & 0xFFFFull) << 48);               // data_size=8B
    unsigned long long qw1 = ((n8 >> 16) & 0xFFFFull) | (1ull << 16) | ((n8 & 0xFFFFull) << 48);
    unsigned long long qw2 = ((n8 & 0xFFFFFFFFull) << 32);
    unsigned long long qw3 = ((n8 >> 32) & 0xFFFFull);
    int32x8 g1;
    g1[0] = (int)(qw0 & 0xFFFFFFFFull); g1[1] = (int)(qw0 >> 32);
    g1[2] = (int)(qw1 & 0xFFFFFFFFull); g1[3] = (int)(qw1 >> 32);
    g1[4] = (int)(qw2 & 0xFFFFFFFFull); g1[5] = (int)(qw2 >> 32);
    g1[6] = (int)(qw3 & 0xFFFFFFFFull); g1[7] = (int)(qw3 >> 32);
    int32x4 gz = {0, 0, 0, 0};
#if __clang_major__ >= 23
    int32x8 gz8 = {0, 0, 0, 0, 0, 0, 0, 0};
    __builtin_amdgcn_tensor_load_to_lds(g0, g1, gz, gz, gz8, 0);
#else
    __builtin_amdgcn_tensor_load_to_lds(g0, g1, gz, gz, 0);
#endif
    __builtin_amdgcn_s_wait_tensorcnt(0);
  }
#else
  for (unsigned int i = threadIdx.x; i < 4 * NMS_PRE_N; i += blockDim.x)
    (&bxs[0][0])[i] = gbx[i];
#endif

#if USE_ASYNC_LDS
  {
    unsigned int i = threadIdx.x;
    if (i < NMS_PRE_N) {
      __builtin_amdgcn_global_load_async_to_lds_b32(
          (GAS int*)(ws_tops + (size_t)b * NMS_PRE_N + i),
          (LAS int*)(&tops[i]), 0, 0);
    }
    __builtin_amdgcn_s_wait_asynccnt(0);
  }
#else
  for (unsigned int i = threadIdx.x; i < NMS_PRE_N; i += blockDim.x)
    tops[i] = ws_tops[(size_t)b * NMS_PRE_N + i];
#endif
  for (unsigned int i = threadIdx.x; i < NMS_PRE_N; i += blockDim.x) {
    supp[i] = 0; keep[i] = 0;
  }
  __syncthreads();
  for (unsigned int i = threadIdx.x; i < NMS_PRE_N; i += blockDim.x) {
    float w = bxs[2][i] - bxs[0][i]; if (w < 0.f) w = 0.f;
    float h = bxs[3][i] - bxs[1][i]; if (h < 0.f) h = 0.f;
    area[i] = w * h;
  }
  __syncthreads();

  // Greedy NMS: serial over i, parallel (1000-wide, SoA conflict-free) per step.
  for (int i = 0; i < NMS_PRE_N; ++i) {
    if (threadIdx.x == 0) {
      int ki = (tops[i] > MIN_SCORE_F) && (supp[i] == 0);
      keep[i] = (unsigned char)ki;
      sh_keep = ki;
    }
    __syncthreads();
    if (sh_keep) {
      const float x1i = bxs[0][i], y1i = bxs[1][i], x2i = bxs[2][i], y2i = bxs[3][i];
      const float ai = area[i];
      const unsigned int j = threadIdx.x;
      if (j < NMS_PRE_N) {
        float xx1 = fmaxf(x1i, bxs[0][j]);
        float yy1 = fmaxf(y1i, bxs[1][j]);
        float xx2 = fminf(x2i, bxs[2][j]);
        float yy2 = fminf(y2i, bxs[3][j]);
        float iw = xx2 - xx1; if (iw < 0.f) iw = 0.f;
        float ih = yy2 - yy1; if (ih < 0.f) ih = 0.f;
        float inter = iw * ih;
        float uni = ai + area[j] - inter;
        if (inter / fmaxf(uni, 1e-6f) > NMS_THR_F) supp[j] = 1;
      }
    }
    __syncthreads();
  }

  // Final stable top-100 of s2 = keep ? score : -1
  for (unsigned int i = threadIdx.x; i < KPAD; i += blockDim.x) {
    unsigned long long k;
    if (i < NMS_PRE_N) {
      float s2 = keep[i] ? tops[i] : -1.0f;
      k = ((unsigned long long)f32_to_dkey(s2) << 32) | (unsigned long long)i;
    } else {
      k = ~0ull;
    }
    keys[i] = k;
  }
  __syncthreads();
  bitonic_sort_1024(keys);

  float* out_b = out;
  float* out_s = out + (size_t)B_IMG * NUM_SELECT_N * 4;
  float* out_l = out_s + (size_t)B_IMG * NUM_SELECT_N;
  for (unsigned int i = threadIdx.x; i < NUM_SELECT_N; i += blockDim.x) {
    unsigned long long k = keys[i];
    unsigned int idx = (unsigned int)(k & 0xFFFFFFFFull);
    float val = dkey_to_f32((unsigned int)(k >> 32));
    size_t o = (size_t)b * NUM_SELECT_N + i;
    out_s[o] = val;
    out_l[o] = (float)ws_label[(size_t)b * NMS_PRE_N + idx];
    out_b[o * 4 + 0] = bxs[0][idx];
    out_b[o * 4 + 1] = bxs[1][idx];
    out_b[o * 4 + 2] = bxs[2][idx];
    out_b[o * 4 + 3] = bxs[3][idx];
  }
}

extern "C" void kernel_launch(void* const* d_in, const int* in_sizes, int n_in,
                              void* d_out, int out_size, void* d_ws, size_t ws_size,
                              hipStream_t stream) {
  (void)in_sizes; (void)n_in; (void)out_size;
  const float* logits = (const float*)d_in[0];   // (64,900,91) f32
  const float* boxes  = (const float*)d_in[1];   // (64,900,4)  f32
  const int*   tsz    = (const int*)d_in[2];     // (64,2)      i32
  float* out = (float*)d_out;                    // 25600 + 6400 + 6400 f32

  char* ws = (char*)d_ws;
  const size_t tops_b  = (size_t)B_IMG * NMS_PRE_N * 4;       // 256000
  const size_t label_b = (size_t)B_IMG * NMS_PRE_N * 4;       // 256000
  const size_t boxes_b = (size_t)B_IMG * 4 * NMS_PRE_N * 4;   // 1024000
  const size_t small_b = tops_b + label_b + boxes_b;          // 1536000
  const size_t keys_b  = (size_t)B_IMG * QC_NUM * 4;          // 20966400

  float* ws_tops  = (float*)ws;
  int*   ws_label = (int*)(ws + tops_b);
  float* ws_boxes = (float*)(ws + tops_b + label_b);
  unsigned int* ws_keys = (unsigned int*)(ws + small_b);

  const bool have_keys = (ws_size >= small_b + keys_b);
  if (have_keys) {
    score_key_kernel<<<2048, 256, 0, stream>>>(logits, ws_keys, B_IMG * QC_NUM);
    topk_select_kernel<true><<<B_IMG, 256, 0, stream>>>(logits, ws_keys, boxes, tsz,
                                                        ws_tops, ws_label, ws_boxes);
  } else {
    topk_select_kernel<false><<<B_IMG, 256, 0, stream>>>(logits, ws_keys, boxes, tsz,
                                                         ws_tops, ws_label, ws_boxes);
  }
  nms_finalize_kernel<<<B_IMG, 1024, 0, stream>>>(ws_tops, ws_label, ws_boxes, out);
}